// SGC_12695923327569
// MI455X (gfx1250) — compile-verified
//
#include <hip/hip_runtime.h>
#include <hip/hip_bf16.h>

#define N_NODES 50000
#define N_EDGES 800000
#define IN_DIM  100
#define OUT_DIM 40
#define N_PAD   48          // OUT_DIM padded to 3 tiles of 16
#define H_ELEMS (N_NODES * IN_DIM)   // 5,000,000

typedef __attribute__((ext_vector_type(2))) float v2f;
typedef __attribute__((ext_vector_type(8))) float v8f;

// ---------------- utility ----------------
__global__ void sgc_zero_f32(float* __restrict__ p, int n) {
    int i = blockIdx.x * blockDim.x + threadIdx.x;
    if (i < n) p[i] = 0.0f;
}

// ---------------- degree / norm ----------------
__global__ void sgc_degree(const int* __restrict__ dst, float* __restrict__ deg) {
    int e = blockIdx.x * blockDim.x + threadIdx.x;
    if (e < N_EDGES) atomicAdd(&deg[dst[e]], 1.0f);
}

__global__ void sgc_norm(float* __restrict__ deg_to_norm) {
    int i = blockIdx.x * blockDim.x + threadIdx.x;
    if (i < N_NODES) {
        float d = deg_to_norm[i];
        deg_to_norm[i] = rsqrtf(fmaxf(d, 1.0f));
    }
}

// ---------------- SpMM scatter: h_out[dst] += h_in[src] * norm[src] ----------------
// one thread = (edge, 4-feature chunk); IN_DIM=100 -> 25 chunks
__global__ void sgc_scatter(const float* __restrict__ h_in,
                            const float* __restrict__ norm,
                            const int* __restrict__ src,
                            const int* __restrict__ dst,
                            float* __restrict__ h_out) {
    int gid = blockIdx.x * blockDim.x + threadIdx.x;
    const int total = N_EDGES * (IN_DIM / 4);
    if (gid >= total) return;
    int e  = gid / (IN_DIM / 4);
    int k0 = (gid - e * (IN_DIM / 4)) * 4;
    int s = src[e];
    int d = dst[e];
    float sc = norm[s];
    const float4 v = *reinterpret_cast<const float4*>(h_in + (size_t)s * IN_DIM + k0);
    float* outp = h_out + (size_t)d * IN_DIM + k0;
    atomicAdd(outp + 0, v.x * sc);
    atomicAdd(outp + 1, v.y * sc);
    atomicAdd(outp + 2, v.z * sc);
    atomicAdd(outp + 3, v.w * sc);
}

// ---------------- post-hop row scale: h[row] *= norm[row] ----------------
__global__ void sgc_rowscale(float* __restrict__ h, const float* __restrict__ norm) {
    int row = blockIdx.x;
    int f = threadIdx.x;
    if (f < IN_DIM) h[(size_t)row * IN_DIM + f] *= norm[row];
}

// ---------------- per-column mean / inv-std (ddof=1) ----------------
__global__ void sgc_colstats(const float* __restrict__ h,
                             float* __restrict__ mean,
                             float* __restrict__ invstd) {
    __shared__ float ssum[256];
    __shared__ float ssq[256];
    int c = blockIdx.x;           // 0..IN_DIM-1
    int t = threadIdx.x;
    float s = 0.0f, q = 0.0f;
    for (int r = t; r < N_NODES; r += 256) {
        float v = h[(size_t)r * IN_DIM + c];
        s += v;
        q += v * v;
    }
    ssum[t] = s; ssq[t] = q;
    __syncthreads();
    for (int w = 128; w > 0; w >>= 1) {
        if (t < w) { ssum[t] += ssum[t + w]; ssq[t] += ssq[t + w]; }
        __syncthreads();
    }
    if (t == 0) {
        float m = ssum[0] / (float)N_NODES;
        float var = (ssq[0] - (float)N_NODES * m * m) / (float)(N_NODES - 1);
        mean[c] = m;
        invstd[c] = rsqrtf(var);
    }
}

// ---------------- fold normalization into weights ----------------
// Wp[o][k] = W[o][k]*invstd[k] (o<OUT_DIM, else 0); cvec[o] = b[o] - sum_k mean[k]*Wp[o][k]
__global__ void sgc_prep(const float* __restrict__ W, const float* __restrict__ b,
                         const float* __restrict__ mean, const float* __restrict__ invstd,
                         float* __restrict__ Wp, float* __restrict__ cvec) {
    __shared__ float red[128];
    int o = blockIdx.x;           // 0..N_PAD-1
    int t = threadIdx.x;          // 128 threads
    float partial = 0.0f;
    if (t < IN_DIM) {
        float wp = (o < OUT_DIM) ? W[o * IN_DIM + t] * invstd[t] : 0.0f;
        Wp[o * IN_DIM + t] = wp;
        partial = mean[t] * wp;
    }
    red[t] = partial;
    __syncthreads();
    for (int w = 64; w > 0; w >>= 1) {
        if (t < w) red[t] += red[t + w];
        __syncthreads();
    }
    if (t == 0) cvec[o] = ((o < OUT_DIM) ? b[o] : 0.0f) - red[0];
}

// ---------------- WMMA GEMM: out = h @ Wp^T + cvec (fp32, 16x16x4) ----------------
// Per wave: one 16(M) x 16(N) output tile. K loop of 25 steps of 4.
// A fragment (16x4 f32): lanes 0-15 hold row M=lane, K={kk,kk+1}; lanes 16-31 row M=lane-16, K={kk+2,kk+3}
// B fragment (4x16 f32): lanes 0-15 hold col N=lane, K={kk,kk+1}; lanes 16-31 col N=lane-16, K={kk+2,kk+3}
// D: VGPR v: lanes 0-15 -> M=v, N=lane; lanes 16-31 -> M=v+8, N=lane-16
__global__ void sgc_wmma_gemm(const float* __restrict__ h,
                              const float* __restrict__ Wp,
                              const float* __restrict__ cvec,
                              float* __restrict__ out) {
    const int MT = N_NODES / 16;          // 3125
    const int NT = N_PAD / 16;            // 3
    const int ntiles = MT * NT;           // 9375
    int wave = threadIdx.x >> 5;          // 8 waves / block (wave32)
    int lane = threadIdx.x & 31;
    int tile = blockIdx.x * 8 + wave;
    if (tile >= ntiles) return;           // whole-wave uniform branch: EXEC stays all-1s
    int mt = tile / NT;
    int nt = tile - mt * NT;
    int mbase = mt * 16;
    int nbase = nt * 16;

    int l16  = lane & 15;
    int kadd = (lane >> 4) << 1;          // 0 for lanes 0-15, 2 for lanes 16-31

    const float* aptr = h  + (size_t)(mbase + l16) * IN_DIM + kadd;
    const float* bptr = Wp + (size_t)(nbase + l16) * IN_DIM + kadd;

    v8f acc = {};
    #pragma unroll 5
    for (int kk = 0; kk < IN_DIM; kk += 4) {
        v2f a, bb;
        a.x  = aptr[kk];  a.y  = aptr[kk + 1];
        bb.x = bptr[kk];  bb.y = bptr[kk + 1];
        // (neg_a, A, neg_b, B, c_mod, C, reuse_a, reuse_b)
        acc = __builtin_amdgcn_wmma_f32_16x16x4_f32(
            false, a, false, bb, (short)0, acc, false, false);
    }

    int col = nbase + l16;
    if (col < OUT_DIM) {                  // divergence only after all WMMAs
        float cb = cvec[col];
        int mofs = mbase + ((lane >> 4) << 3);   // +8 for upper half-lanes
        #pragma unroll
        for (int v = 0; v < 8; ++v) {
            out[(size_t)(mofs + v) * OUT_DIM + col] = acc[v] + cb;
        }
    }
}

// ---------------- launch ----------------
extern "C" void kernel_launch(void* const* d_in, const int* in_sizes, int n_in,
                              void* d_out, int out_size, void* d_ws, size_t ws_size,
                              hipStream_t stream) {
    const float* feat = (const float*)d_in[0];   // [N_NODES, IN_DIM]
    const float* W    = (const float*)d_in[1];   // [OUT_DIM, IN_DIM]
    const float* b    = (const float*)d_in[2];   // [OUT_DIM]
    const int*   src  = (const int*)d_in[3];     // [N_EDGES]
    const int*   dst  = (const int*)d_in[4];     // [N_EDGES]
    float* outp = (float*)d_out;                 // [N_NODES, OUT_DIM]

    float* ws     = (float*)d_ws;
    float* norm   = ws;                          // N_NODES
    float* hA     = norm + N_NODES;              // H_ELEMS
    float* hB     = hA + H_ELEMS;                // H_ELEMS
    float* Wp     = hB + H_ELEMS;                // N_PAD*IN_DIM
    float* cvec   = Wp + N_PAD * IN_DIM;         // N_PAD
    float* mean   = cvec + N_PAD;                // 128
    float* invstd = mean + 128;                  // 128

    const int B256 = 256;
    // degree -> norm
    sgc_zero_f32<<<(N_NODES + B256 - 1) / B256, B256, 0, stream>>>(norm, N_NODES);
    sgc_degree<<<(N_EDGES + B256 - 1) / B256, B256, 0, stream>>>(dst, norm);
    sgc_norm<<<(N_NODES + B256 - 1) / B256, B256, 0, stream>>>(norm);

    const int scat_total  = N_EDGES * (IN_DIM / 4);
    const int scat_blocks = (scat_total + B256 - 1) / B256;
    const int zero_blocks = (H_ELEMS + B256 - 1) / B256;

    // hop 0: feat -> hA
    sgc_zero_f32<<<zero_blocks, B256, 0, stream>>>(hA, H_ELEMS);
    sgc_scatter<<<scat_blocks, B256, 0, stream>>>(feat, norm, src, dst, hA);
    sgc_rowscale<<<N_NODES, 128, 0, stream>>>(hA, norm);
    // hop 1: hA -> hB
    sgc_zero_f32<<<zero_blocks, B256, 0, stream>>>(hB, H_ELEMS);
    sgc_scatter<<<scat_blocks, B256, 0, stream>>>(hA, norm, src, dst, hB);
    sgc_rowscale<<<N_NODES, 128, 0, stream>>>(hB, norm);
    // hop 2: hB -> hA
    sgc_zero_f32<<<zero_blocks, B256, 0, stream>>>(hA, H_ELEMS);
    sgc_scatter<<<scat_blocks, B256, 0, stream>>>(hB, norm, src, dst, hA);
    sgc_rowscale<<<N_NODES, 128, 0, stream>>>(hA, norm);

    // column standardization folded into classifier weights
    sgc_colstats<<<IN_DIM, 256, 0, stream>>>(hA, mean, invstd);
    sgc_prep<<<N_PAD, 128, 0, stream>>>(W, b, mean, invstd, Wp, cvec);

    // fused normalize + linear via fp32 WMMA
    const int ntiles = (N_NODES / 16) * (N_PAD / 16);   // 9375 wave-tiles
    const int gblocks = (ntiles + 7) / 8;               // 8 waves per 256-thread block
    sgc_wmma_gemm<<<gblocks, B256, 0, stream>>>(hA, Wp, cvec, outp);
}